// SpatioTemporalGNN_790273982716
// MI455X (gfx1250) — compile-verified
//
#include <hip/hip_runtime.h>
#include <hip/hip_bf16.h>
#include <math.h>

// ---------------------------------------------------------------------------
// SpatioTemporalGNN forward for MI455X (gfx1250, wave32).
// Dense GEMMs use V_WMMA_F32_16X16X4_F32 (full fp32 accuracy, matrix pipe),
// with B staged in LDS (K-pair interleaved so each fragment is one b64 read)
// and 4-way N register blocking per wave.
// Edge softmax-aggregation uses fp32 atomics (bandwidth-bound part).
// ---------------------------------------------------------------------------

typedef __attribute__((ext_vector_type(2))) float v2f;
typedef __attribute__((ext_vector_type(8))) float v8f;

#define HIDC   64
#define NHEAD  4
#define GDIM   256   // NHEAD * HIDC

__device__ __forceinline__ float elu_f(float x) { return x > 0.0f ? x : expm1f(x); }

// sign-aware float atomic max (works for mixed-sign values)
__device__ __forceinline__ void atomicMaxFloat(float* addr, float value) {
    if (value >= 0.0f)
        atomicMax((int*)addr, __float_as_int(value));
    else
        atomicMin((unsigned int*)addr, __float_as_uint(value));
}

// ---------------------------------------------------------------------------
// fp32 WMMA GEMM: C[M,Ncols] = act(A[M,K] @ B[K,Ncols] + bias)
// Requirements: K % 4 == 0, K <= 64, Ncols % 64 == 0.
// Block = 256 threads = 8 waves. Each block handles 8 consecutive M-tiles for
// one 64-column group; the Kx64 B tile is staged in LDS once per block with
// K-pair interleaving: element (k,col) -> Bs[(k>>1)*128 + col*2 + (k&1)], so
// the per-WMMA B fragment {B[ka][col], B[ka+1][col]} is a single ds_load_b64.
// Each wave computes a 16x64 slab: one A fragment feeds 4 WMMAs per K-step.
// A layout (16x4 f32): lanes 0-15 -> K={k,k+1}, lanes 16-31 -> K={k+2,k+3}.
// B layout (4x16 f32): VGPR0 = rows k (low lanes) / k+2 (high lanes),
//                      VGPR1 = rows k+1 / k+3.
// C/D layout: VGPR i -> row i (lanes 0-15) and row i+8 (lanes 16-31).
// ---------------------------------------------------------------------------
__global__ void gemm_f32_wmma(const float* __restrict__ A, const float* __restrict__ B,
                              const float* __restrict__ bias, float* __restrict__ C,
                              int M, int K, int Ncols, int act) {
    __shared__ float Bs[64 * 64];                       // up to K=64, 64 cols (16 KB)

    const int ncg_count = Ncols >> 6;                   // 64-col groups
    const int mtiles    = (M + 15) >> 4;
    const int mblk      = blockIdx.x / ncg_count;       // 8 m-tiles per block
    const int ncg       = blockIdx.x % ncg_count;
    const int colbase   = ncg << 6;

    // cooperative stage of B tile (K x 64), K-pair interleaved
    for (int i = threadIdx.x; i < K * 64; i += 256) {
        const int k = i >> 6, col = i & 63;
        Bs[(k >> 1) * 128 + col * 2 + (k & 1)] = B[(size_t)k * Ncols + colbase + col];
    }
    __syncthreads();

    const int w  = threadIdx.x >> 5;                    // wave in block
    const int mt = mblk * 8 + w;
    if (mt >= mtiles) return;                           // wave-uniform, after barrier
    const int lane = threadIdx.x & 31;
    const int half = lane >> 4;
    const int l16  = lane & 15;

    int arow = mt * 16 + l16;
    if (arow >= M) arow = M - 1;                        // clamp loads; stores guarded

    v8f c0 = {}, c1 = {}, c2 = {}, c3 = {};
    for (int k0 = 0; k0 < K; k0 += 4) {
        const int ka = k0 + half * 2;                   // even
        v2f a = *(const v2f*)&A[(size_t)arow * K + ka];
        const float* bp = &Bs[(ka >> 1) * 128 + l16 * 2];
        v2f b0 = *(const v2f*)&bp[0];
        v2f b1 = *(const v2f*)&bp[32];
        v2f b2 = *(const v2f*)&bp[64];
        v2f b3 = *(const v2f*)&bp[96];
        c0 = __builtin_amdgcn_wmma_f32_16x16x4_f32(false, a, false, b0, (short)0, c0, false, false);
        c1 = __builtin_amdgcn_wmma_f32_16x16x4_f32(false, a, false, b1, (short)0, c1, false, false);
        c2 = __builtin_amdgcn_wmma_f32_16x16x4_f32(false, a, false, b2, (short)0, c2, false, false);
        c3 = __builtin_amdgcn_wmma_f32_16x16x4_f32(false, a, false, b3, (short)0, c3, false, false);
    }

#pragma unroll
    for (int j = 0; j < 4; ++j) {
        const v8f c = (j == 0) ? c0 : (j == 1) ? c1 : (j == 2) ? c2 : c3;
        const int col = colbase + j * 16 + l16;
#pragma unroll
        for (int i = 0; i < 8; ++i) {
            const int row = mt * 16 + i + half * 8;
            if (row < M) {
                float v = c[i];
                if (bias) v += bias[col];
                if (act == 1) v = elu_f(v);
                C[(size_t)row * Ncols + col] = v;
            }
        }
    }
}

static inline int gemm_blocks(int M, int Ncols) {
    const int mtiles = (M + 15) / 16;
    return ((mtiles + 7) / 8) * (Ncols / 64);
}

// ---------------------------------------------------------------------------
__global__ void fill_kernel(float* __restrict__ p, float v, size_t n) {
    size_t i = (size_t)blockIdx.x * blockDim.x + threadIdx.x;
    if (i < n) p[i] = v;
}

// s_src[n,h] = sum_c g[n,h,c]*a_src[h,c]   (and s_dst)
__global__ void score_kernel(const float* __restrict__ g,
                             const float* __restrict__ a_src,
                             const float* __restrict__ a_dst,
                             float* __restrict__ ssrc, float* __restrict__ sdst, int Nn) {
    int i = blockIdx.x * blockDim.x + threadIdx.x;
    if (i >= Nn * NHEAD) return;
    int n = i >> 2, h = i & 3;
    const float* gp = g + (size_t)n * GDIM + h * HIDC;
    const float* as = a_src + h * HIDC;
    const float* ad = a_dst + h * HIDC;
    float s1 = 0.f, s2 = 0.f;
    for (int c = 0; c < HIDC; ++c) { float v = gp[c]; s1 += v * as[c]; s2 += v * ad[c]; }
    ssrc[i] = s1;
    sdst[i] = s2;
}

// edge table: first E entries come from edge_index, then Nn self-loops
__device__ __forceinline__ void edge_info(int e, int E, const int* __restrict__ ei,
                                          const int* __restrict__ et, int btype,
                                          int& s, int& d, bool& keep) {
    if (e < E) {
        s = ei[e];
        d = ei[E + e];
        keep = (et[e] == btype) && (s != d);
    } else {
        s = e - E;
        d = s;
        keep = true;
    }
}

// pass1: score = leaky_relu(s_src[src]+s_dst[dst]); mask; segment max over dst
__global__ void edge_score_max(const int* __restrict__ ei, const int* __restrict__ et,
                               int E, int Nn, int btype,
                               const float* __restrict__ ssrc, const float* __restrict__ sdst,
                               float* __restrict__ esc, float* __restrict__ smax) {
    int e = blockIdx.x * blockDim.x + threadIdx.x;
    if (e >= E + Nn) return;
    int s, d; bool keep;
    edge_info(e, E, ei, et, btype, s, d, keep);
#pragma unroll
    for (int h = 0; h < NHEAD; ++h) {
        float sc = ssrc[s * NHEAD + h] + sdst[d * NHEAD + h];
        sc = sc > 0.0f ? sc : 0.2f * sc;          // leaky_relu, slope 0.2
        if (!keep) sc = -1e9f;
        esc[(size_t)e * NHEAD + h] = sc;
        atomicMaxFloat(&smax[d * NHEAD + h], sc);
    }
}

// pass2: ex = exp(score - smax[dst]); segment sum over dst
__global__ void edge_exp_sum(const int* __restrict__ ei, const int* __restrict__ et,
                             int E, int Nn, int btype,
                             float* __restrict__ esc, const float* __restrict__ smax,
                             float* __restrict__ denom) {
    int e = blockIdx.x * blockDim.x + threadIdx.x;
    if (e >= E + Nn) return;
    int s, d; bool keep;
    edge_info(e, E, ei, et, btype, s, d, keep);
#pragma unroll
    for (int h = 0; h < NHEAD; ++h) {
        float ex = expf(esc[(size_t)e * NHEAD + h] - smax[d * NHEAD + h]);
        esc[(size_t)e * NHEAD + h] = ex;
        atomicAdd(&denom[d * NHEAD + h], ex);
    }
}

// pass3: acc[dst,h,c] += (ex/denom) * g[src,h,c]; one 256-thread block per edge
__global__ void edge_aggregate(const int* __restrict__ ei, const int* __restrict__ et,
                               int E, int Nn, int btype,
                               const float* __restrict__ esc, const float* __restrict__ denom,
                               const float* __restrict__ g, float* __restrict__ acc) {
    int e = blockIdx.x;
    int t = threadIdx.x;          // 256 threads: (head, channel)
    int h = t >> 6;
    int c = t & 63;
    int s, d; bool keep;
    edge_info(e, E, ei, et, btype, s, d, keep);
    float ex = esc[(size_t)e * NHEAD + h];
    if (ex == 0.0f) return;       // masked / underflowed edges contribute nothing
    float alpha = ex / (denom[d * NHEAD + h] + 1e-16f);
    float v = alpha * g[(size_t)s * GDIM + h * HIDC + c];
    atomicAdd(&acc[(size_t)d * GDIM + h * HIDC + c], v);
}

// mean over heads + bias, then LayerNorm + ELU.  64 threads per node.
__global__ void gat_finalize(const float* __restrict__ acc, const float* __restrict__ bias,
                             const float* __restrict__ ln_g, const float* __restrict__ ln_b,
                             float* __restrict__ out, int Nn) {
    int n = blockIdx.x;
    int c = threadIdx.x;                    // 64
    __shared__ float red[HIDC];
    size_t base = (size_t)n * GDIM;
    float v = 0.25f * (acc[base + c] + acc[base + 64 + c] +
                       acc[base + 128 + c] + acc[base + 192 + c]) + bias[c];
    red[c] = v;
    __syncthreads();
    for (int s = 32; s > 0; s >>= 1) { if (c < s) red[c] += red[c + s]; __syncthreads(); }
    float mu = red[0] * (1.0f / 64.0f);
    __syncthreads();
    float dv = v - mu;
    red[c] = dv * dv;
    __syncthreads();
    for (int s = 32; s > 0; s >>= 1) { if (c < s) red[c] += red[c + s]; __syncthreads(); }
    float var = red[0] * (1.0f / 64.0f);
    float y = dv * rsqrtf(var + 1e-5f) * ln_g[c] + ln_b[c];
    out[(size_t)n * HIDC + c] = elu_f(y);
}

// fused head: gate = sigmoid([hs|ht]@gW + gb); hf = gate*hs+(1-gate)*ht;
// z = relu(hf@W1+b1); logits = z@W2+b2.  64 threads per node.
__global__ void head_kernel(const float* __restrict__ hs, const float* __restrict__ ht,
                            const float* __restrict__ gW, const float* __restrict__ gb,
                            const float* __restrict__ W1, const float* __restrict__ b1,
                            const float* __restrict__ W2, const float* __restrict__ b2,
                            float* __restrict__ logits, float* __restrict__ hf_out, int Nn) {
    int n = blockIdx.x;
    int j = threadIdx.x;                    // 64
    __shared__ float cat[2 * HIDC];
    __shared__ float hf[HIDC];
    __shared__ float z[32];
    cat[j]        = hs[(size_t)n * HIDC + j];
    cat[HIDC + j] = ht[(size_t)n * HIDC + j];
    __syncthreads();
    float a = gb[j];
    for (int k = 0; k < 2 * HIDC; ++k) a += cat[k] * gW[k * HIDC + j];
    float gate = 1.0f / (1.0f + expf(-a));
    float f = gate * cat[j] + (1.0f - gate) * cat[HIDC + j];
    hf[j] = f;
    hf_out[(size_t)n * HIDC + j] = f;
    __syncthreads();
    if (j < 32) {
        float s = b1[j];
        for (int k = 0; k < HIDC; ++k) s += hf[k] * W1[k * 32 + j];
        z[j] = s > 0.0f ? s : 0.0f;
    }
    __syncthreads();
    if (j < 8) {
        float s = b2[j];
        for (int k = 0; k < 32; ++k) s += z[k] * W2[k * 8 + j];
        logits[(size_t)n * 8 + j] = s;
    }
}

// ---------------------------------------------------------------------------
// Host orchestration
// ---------------------------------------------------------------------------
static void run_gat(const float* h_in, const float* W, const float* a_src,
                    const float* a_dst, const float* bgat,
                    const float* ln_g, const float* ln_b,
                    const int* ei, const int* et, int btype, int Nn, int E,
                    float* g, float* acc, float* ssrc, float* sdst,
                    float* smax, float* denom, float* esc, float* h_out,
                    hipStream_t stream) {
    const int Etot = E + Nn;
    // g = h_in @ W   (M=Nn, K=64, Ncols=256)
    gemm_f32_wmma<<<gemm_blocks(Nn, GDIM), 256, 0, stream>>>(h_in, W, nullptr, g,
                                                             Nn, HIDC, GDIM, 0);
    score_kernel<<<(Nn * NHEAD + 255) / 256, 256, 0, stream>>>(g, a_src, a_dst,
                                                               ssrc, sdst, Nn);
    fill_kernel<<<((size_t)Nn * NHEAD + 255) / 256, 256, 0, stream>>>(smax, -3.0e38f,
                                                                      (size_t)Nn * NHEAD);
    fill_kernel<<<((size_t)Nn * NHEAD + 255) / 256, 256, 0, stream>>>(denom, 0.0f,
                                                                      (size_t)Nn * NHEAD);
    fill_kernel<<<((size_t)Nn * GDIM + 255) / 256, 256, 0, stream>>>(acc, 0.0f,
                                                                     (size_t)Nn * GDIM);
    edge_score_max<<<(Etot + 255) / 256, 256, 0, stream>>>(ei, et, E, Nn, btype,
                                                           ssrc, sdst, esc, smax);
    edge_exp_sum<<<(Etot + 255) / 256, 256, 0, stream>>>(ei, et, E, Nn, btype,
                                                         esc, smax, denom);
    edge_aggregate<<<Etot, 256, 0, stream>>>(ei, et, E, Nn, btype, esc, denom, g, acc);
    gat_finalize<<<Nn, 64, 0, stream>>>(acc, bgat, ln_g, ln_b, h_out, Nn);
}

extern "C" void kernel_launch(void* const* d_in, const int* in_sizes, int n_in,
                              void* d_out, int out_size, void* d_ws, size_t ws_size,
                              hipStream_t stream) {
    const float* x  = (const float*)d_in[0];
    const int*   ei = (const int*)d_in[1];     // [2,E] flat: src row then dst row
    const int*   et = (const int*)d_in[2];     // [E]
    const int Nn = in_sizes[0] / 32;
    const int E  = in_sizes[2];

    // params dict leaves in insertion order
    const float* P[35];
    for (int i = 3; i < n_in && i < 35; ++i) P[i] = (const float*)d_in[i];
    const float* in_W = P[3];  const float* in_b = P[4];
    // branch s
    const float *s1W = P[5],  *s1as = P[6],  *s1ad = P[7],  *s1b = P[8];
    const float *s2W = P[9],  *s2as = P[10], *s2ad = P[11], *s2b = P[12];
    const float *sn1g = P[13], *sn1b = P[14], *sn2g = P[15], *sn2b = P[16];
    // branch t
    const float *t1W = P[17], *t1as = P[18], *t1ad = P[19], *t1b = P[20];
    const float *t2W = P[21], *t2as = P[22], *t2ad = P[23], *t2b = P[24];
    const float *tn1g = P[25], *tn1b = P[26], *tn2g = P[27], *tn2b = P[28];
    const float *gW = P[29], *gb = P[30];
    const float *W1 = P[31], *b1 = P[32], *W2 = P[33], *b2 = P[34];

    // workspace layout
    float* ws = (float*)d_ws;
    size_t o = 0;
    float* h0   = ws + o; o += (size_t)Nn * HIDC;
    float* g    = ws + o; o += (size_t)Nn * GDIM;
    float* acc  = ws + o; o += (size_t)Nn * GDIM;
    float* ssrc = ws + o; o += (size_t)Nn * NHEAD;
    float* sdst = ws + o; o += (size_t)Nn * NHEAD;
    float* smax = ws + o; o += (size_t)Nn * NHEAD;
    float* den  = ws + o; o += (size_t)Nn * NHEAD;
    float* esc  = ws + o; o += (size_t)(E + Nn) * NHEAD;
    float* htmp = ws + o; o += (size_t)Nn * HIDC;
    float* hs   = ws + o; o += (size_t)Nn * HIDC;
    float* ht   = ws + o; o += (size_t)Nn * HIDC;
    (void)ws_size; (void)out_size;

    // h0 = elu(x @ in_W + in_b)   (M=Nn, K=32, Ncols=64)
    gemm_f32_wmma<<<gemm_blocks(Nn, HIDC), 256, 0, stream>>>(x, in_W, in_b, h0,
                                                             Nn, 32, HIDC, 1);

    // branch s (edge_type == 0)
    run_gat(h0,   s1W, s1as, s1ad, s1b, sn1g, sn1b, ei, et, 0, Nn, E,
            g, acc, ssrc, sdst, smax, den, esc, htmp, stream);
    run_gat(htmp, s2W, s2as, s2ad, s2b, sn2g, sn2b, ei, et, 0, Nn, E,
            g, acc, ssrc, sdst, smax, den, esc, hs, stream);
    // branch t (edge_type == 1)
    run_gat(h0,   t1W, t1as, t1ad, t1b, tn1g, tn1b, ei, et, 1, Nn, E,
            g, acc, ssrc, sdst, smax, den, esc, htmp, stream);
    run_gat(htmp, t2W, t2as, t2ad, t2b, tn2g, tn2b, ei, et, 1, Nn, E,
            g, acc, ssrc, sdst, smax, den, esc, ht, stream);

    // head: logits [Nn,8] then h_fused [Nn,64] into d_out
    float* logits = (float*)d_out;
    float* hf_out = (float*)d_out + (size_t)Nn * 8;
    head_kernel<<<Nn, 64, 0, stream>>>(hs, ht, gW, gb, W1, b1, W2, b2,
                                       logits, hf_out, Nn);
}